// NeighborEmbedding_own_14628658610723
// MI455X (gfx1250) — compile-verified
//
#include <hip/hip_runtime.h>
#include <hip/hip_bf16.h>
#include <math.h>

// ---------------- problem constants ----------------
#define B_    8
#define N_    1620
#define KNN   32
#define BN_   (B_ * N_)          // 12960
#define NT16  ((N_ + 15) / 16)   // 102 column tiles of 16
#define NT64  ((N_ + 63) / 64)   // 26 column tiles of 64

typedef float v2f __attribute__((ext_vector_type(2)));
typedef float v8f __attribute__((ext_vector_type(8)));

// fp32 WMMA: D(16x16) += A(16x4) * B(4x16), exact fp32 on the matrix pipe.
__device__ __forceinline__ v8f wmma_f32(v2f a, v2f b, v8f c) {
  return __builtin_amdgcn_wmma_f32_16x16x4_f32(false, a, false, b, (short)0, c,
                                               false, false);
}

// ---------------- workspace layout (floats) ----------------
// total = 11,964,384 floats = 47.9 MB
#define W_H48   0u
#define W_X0    622080u
#define W_IDX   1244160u
#define W_X1    1658880u
#define W_X2    2903040u
#define W_X3    4147200u
#define W_X4    5391360u
#define W_X5    6635520u
#define W_Q     7879680u
#define W_V     8190720u
#define W_XR    9434880u
#define W_T     10679040u
#define W_RMAX  11923200u
#define W_RSUM  11936160u
#define W_COLS  11949120u
#define W_STATS 11962080u
// stats arena slots (relative to W_STATS)
#define S_BN1 0
#define S_BN2 96
#define S_SG1 192
#define S_SG2 384
#define S_T0  576
#define S_T1  768
#define S_T2  960
#define S_T3  1152
#define S_LIN 1344
#define STATS_FLOATS 2304

// =====================================================================
// conv1: 3 -> 48 (scalar; K=3 too small for WMMA). Writes pre-BN h48.
// =====================================================================
__global__ __launch_bounds__(256) void conv1_kernel(
    const float* __restrict__ x, const float* __restrict__ w,
    float* __restrict__ h48) {
  int p = blockIdx.x * 256 + threadIdx.x;
  int o = blockIdx.y;
  if (p >= BN_) return;
  int b = p / N_, n = p % N_;
  const float* xb = x + (size_t)b * 3 * N_ + n;
  float v = w[o * 3 + 0] * xb[0] + w[o * 3 + 1] * xb[(size_t)N_] +
            w[o * 3 + 2] * xb[(size_t)2 * N_];
  h48[((size_t)b * 48 + o) * N_ + n] = v;
}

// =====================================================================
// Generic channel-major 1x1-conv GEMM via fp32 WMMA, 16x64 per wave.
// Y[b,o,n] = sum_c W[o,c] * X[b,c,n] (+ bias[o]).
// A fragment (weights) is reused across 4 column tiles -> ~2.25 VMEM/WMMA.
// grid: (ceil(N/64), ceil(OC/16), B), block = 32 (one wave)
// =====================================================================
__global__ __launch_bounds__(32) void gemm_wmma_kernel(
    const float* __restrict__ W, const float* __restrict__ bias,
    const float* __restrict__ X, float* __restrict__ Y, int OC, int IC) {
  const int lane = threadIdx.x & 31;
  const int half = lane >> 4, l16 = lane & 15;
  const int n0 = blockIdx.x * 64;
  const int m0 = blockIdx.y * 16;
  const int b = blockIdx.z;
  const int m = m0 + l16;
  const int mr = (m < OC) ? m : 0;  // clamp A row (masked at store)
  int nr[4];
#pragma unroll
  for (int j = 0; j < 4; j++) {
    int n = n0 + 16 * j + l16;
    nr[j] = (n < N_) ? n : N_ - 1;  // clamp B col (masked at store)
  }
  const float* Xb = X + (size_t)b * IC * N_;
  v8f acc[4] = {};
  for (int k = 0; k < IC; k += 4) {
    int ka = k + 2 * half;
    v2f a;
    a.x = W[(size_t)mr * IC + ka];
    a.y = W[(size_t)mr * IC + ka + 1];
    const float* r0 = Xb + (size_t)ka * N_;
    const float* r1 = Xb + (size_t)(ka + 1) * N_;
#pragma unroll
    for (int j = 0; j < 4; j++) {
      v2f bb;
      bb.x = r0[nr[j]];
      bb.y = r1[nr[j]];
      acc[j] = wmma_f32(a, bb, acc[j]);
    }
  }
  float* Yb = Y + (size_t)b * OC * N_;
#pragma unroll
  for (int j = 0; j < 4; j++) {
    int n = n0 + 16 * j + l16;
    if (n < N_) {
#pragma unroll
      for (int r = 0; r < 8; r++) {
        int mm = m0 + r + 8 * half;
        if (mm < OC) {
          float v = acc[j][r];
          if (bias) v += bias[mm];
          Yb[(size_t)mm * N_ + n] = v;
        }
      }
    }
  }
}

// =====================================================================
// lin GEMM (480x480) reading the concatenation [x1..x5] without a copy.
// Same 16x64-per-wave shape.  grid: (NT64, 30, B)
// =====================================================================
__global__ __launch_bounds__(32) void gemm_cat_kernel(
    const float* __restrict__ W, const float* __restrict__ x1,
    const float* __restrict__ x2, const float* __restrict__ x3,
    const float* __restrict__ x4, const float* __restrict__ x5,
    float* __restrict__ Y) {
  const float* bufs[5] = {x1, x2, x3, x4, x5};
  const int lane = threadIdx.x & 31;
  const int half = lane >> 4, l16 = lane & 15;
  const int n0 = blockIdx.x * 64;
  const int m0 = blockIdx.y * 16;
  const int b = blockIdx.z;
  int nr[4];
#pragma unroll
  for (int j = 0; j < 4; j++) {
    int n = n0 + 16 * j + l16;
    nr[j] = (n < N_) ? n : N_ - 1;
  }
  v8f acc[4] = {};
  for (int k = 0; k < 480; k += 4) {
    int ka = k + 2 * half;  // even -> ka, ka+1 stay in the same buffer
    const float* Bk = bufs[ka / 96] + ((size_t)b * 96 + (ka % 96)) * N_;
    v2f a;
    a.x = W[(size_t)(m0 + l16) * 480 + ka];
    a.y = W[(size_t)(m0 + l16) * 480 + ka + 1];
#pragma unroll
    for (int j = 0; j < 4; j++) {
      v2f bb;
      bb.x = Bk[nr[j]];
      bb.y = Bk[(size_t)N_ + nr[j]];
      acc[j] = wmma_f32(a, bb, acc[j]);
    }
  }
  float* Yb = Y + (size_t)b * 480 * N_;
#pragma unroll
  for (int j = 0; j < 4; j++) {
    int n = n0 + 16 * j + l16;
    if (n < N_) {
#pragma unroll
      for (int r = 0; r < 8; r++) {
        int mm = m0 + r + 8 * half;
        Yb[(size_t)mm * N_ + n] = acc[j][r];
      }
    }
  }
}

// =====================================================================
// Deterministic per-channel BN statistics: one block per channel.
// buf layout (B, C, N); writes stats[c]=sum, stats[C+c]=sumsq.
// =====================================================================
__global__ __launch_bounds__(256) void stats_kernel(
    const float* __restrict__ buf, float* __restrict__ st, int C) {
  int c = blockIdx.x;
  float s = 0.f, sq = 0.f;
  for (int p = threadIdx.x; p < BN_; p += 256) {
    int b = p / N_, n = p % N_;
    float v = buf[((size_t)b * C + c) * N_ + n];
    s += v;
    sq += v * v;
  }
  __shared__ float rs[256], rq[256];
  rs[threadIdx.x] = s;
  rq[threadIdx.x] = sq;
  __syncthreads();
  for (int o = 128; o > 0; o >>= 1) {
    if (threadIdx.x < o) {
      rs[threadIdx.x] += rs[threadIdx.x + o];
      rq[threadIdx.x] += rq[threadIdx.x + o];
    }
    __syncthreads();
  }
  if (threadIdx.x == 0) {
    st[c] = rs[0];
    st[C + c] = rq[0];
  }
}

__global__ void bn_finalize_kernel(float* st, const float* __restrict__ g,
                                   const float* __restrict__ bsh, int C,
                                   float invcnt) {
  int c = blockIdx.x * blockDim.x + threadIdx.x;
  if (c >= C) return;
  float mean = st[c] * invcnt;
  float var = st[C + c] * invcnt - mean * mean;
  float sc = g[c] * rsqrtf(var + 1e-5f);
  st[c] = sc;                      // scale
  st[C + c] = bsh[c] - sc * mean;  // shift
}

// mode 0: relu(bn)   mode 1: resid + relu(bn)   mode 2: leaky(bn, 0.2)
__global__ __launch_bounds__(256) void bn_apply_kernel(
    const float* __restrict__ in, const float* __restrict__ resid,
    float* __restrict__ out, const float* __restrict__ st, int C, int mode) {
  size_t i = (size_t)blockIdx.x * 256 + threadIdx.x;
  size_t total = (size_t)B_ * C * N_;
  if (i >= total) return;
  int c = (int)((i / N_) % C);
  float v = st[c] * in[i] + st[C + c];
  if (mode == 0)
    v = fmaxf(v, 0.f);
  else if (mode == 1)
    v = resid[i] + fmaxf(v, 0.f);
  else
    v = (v > 0.f) ? v : 0.2f * v;
  out[i] = v;
}

// =====================================================================
// kNN (K=32 smallest sq-distances over xyz).  Neighbor order irrelevant
// downstream (per-neighbor MLP then max over K).
// =====================================================================
__global__ __launch_bounds__(64) void knn_kernel(const float* __restrict__ x,
                                                 int* __restrict__ idxb) {
  __shared__ float sd[64][KNN];
  __shared__ int si[64][KNN];
  int t = threadIdx.x;
  int p = blockIdx.x * 64 + t;
  if (p >= BN_) return;
  int b = p / N_, n = p % N_;
  const float* xb = x + (size_t)b * 3 * N_;
  float px = xb[n], py = xb[(size_t)N_ + n], pz = xb[(size_t)2 * N_ + n];
  float* bd = sd[t];
  int* bi = si[t];
  for (int k = 0; k < KNN; k++) {
    bd[k] = 3.4e38f;
    bi[k] = 0;
  }
  float worst = 3.4e38f;
  int wslot = 0;
  for (int m = 0; m < N_; m++) {
    float dx = px - xb[m];
    float dy = py - xb[(size_t)N_ + m];
    float dz = pz - xb[(size_t)2 * N_ + m];
    float d = dx * dx + dy * dy + dz * dz;
    if (d < worst) {
      bd[wslot] = d;
      bi[wslot] = m;
      worst = -1.f;
      for (int k = 0; k < KNN; k++)
        if (bd[k] > worst) {
          worst = bd[k];
          wslot = k;
        }
    }
  }
  for (int k = 0; k < KNN; k++) idxb[(size_t)p * KNN + k] = bi[k];
}

// =====================================================================
// sample-&-group MLP, fused per point: gather -> W1 -> [stats | bn1+relu
// -> W2 -> [stats | bn2+relu -> max over K]].
// 6 waves/block, each wave does a 16x32 tile (A fragment reused across the
// two 16-column tiles).  All GEMM work on WMMA.  grid = BN_ blocks.
// =====================================================================
__global__ __launch_bounds__(192) void sg_kernel(
    const float* __restrict__ x0, const int* __restrict__ idxb,
    const float* __restrict__ w1, const float* __restrict__ w2,
    const float* __restrict__ ss1, const float* __restrict__ ss2,
    float* __restrict__ stats_out, float* __restrict__ x1out, int mode) {
  __shared__ float nf[96][KNN];
  __shared__ float h1[96][KNN];
  __shared__ float cen[48];
  __shared__ int nidx[KNN];
  __shared__ float sacc[192];
  __shared__ float rmax[96];

  int p = blockIdx.x;
  int b = p / N_, n = p % N_;
  int t = threadIdx.x;
  if (t < KNN) nidx[t] = idxb[(size_t)p * KNN + t];
  if (t < 48) cen[t] = x0[((size_t)b * 48 + t) * N_ + n];
  sacc[t] = 0.f;
  __syncthreads();
  for (int e = t; e < 96 * KNN; e += 192) {
    int c = e / KNN, k = e % KNN;
    float v;
    if (c < 48)
      v = x0[((size_t)b * 48 + c) * N_ + nidx[k]] - cen[c];
    else
      v = cen[c - 48];
    nf[c][k] = v;
  }
  __syncthreads();

  int wave = t >> 5, lane = t & 31, half = lane >> 4, l16 = lane & 15;
  int m0 = wave * 16;

  v8f a0 = {}, a1 = {};
  for (int k = 0; k < 96; k += 4) {
    int ka = k + 2 * half;
    v2f a, b0, b1;
    a.x = w1[(size_t)(m0 + l16) * 96 + ka];
    a.y = w1[(size_t)(m0 + l16) * 96 + ka + 1];
    b0.x = nf[ka][l16];
    b0.y = nf[ka + 1][l16];
    b1.x = nf[ka][16 + l16];
    b1.y = nf[ka + 1][16 + l16];
    a0 = wmma_f32(a, b0, a0);
    a1 = wmma_f32(a, b1, a1);
  }

  if (mode == 0) {
#pragma unroll
    for (int r = 0; r < 8; r++) {
      int c = m0 + r + 8 * half;
      atomicAdd(&sacc[c], a0[r] + a1[r]);
      atomicAdd(&sacc[96 + c], a0[r] * a0[r] + a1[r] * a1[r]);
    }
    __syncthreads();
    atomicAdd(&stats_out[t], sacc[t]);
    return;
  }

#pragma unroll
  for (int r = 0; r < 8; r++) {
    int c = m0 + r + 8 * half;
    h1[c][l16] = fmaxf(ss1[c] * a0[r] + ss1[96 + c], 0.f);
    h1[c][16 + l16] = fmaxf(ss1[c] * a1[r] + ss1[96 + c], 0.f);
  }
  __syncthreads();

  v8f c0 = {}, c1 = {};
  for (int k = 0; k < 96; k += 4) {
    int ka = k + 2 * half;
    v2f a, b0, b1;
    a.x = w2[(size_t)(m0 + l16) * 96 + ka];
    a.y = w2[(size_t)(m0 + l16) * 96 + ka + 1];
    b0.x = h1[ka][l16];
    b0.y = h1[ka + 1][l16];
    b1.x = h1[ka][16 + l16];
    b1.y = h1[ka + 1][16 + l16];
    c0 = wmma_f32(a, b0, c0);
    c1 = wmma_f32(a, b1, c1);
  }

  if (mode == 1) {
#pragma unroll
    for (int r = 0; r < 8; r++) {
      int c = m0 + r + 8 * half;
      atomicAdd(&sacc[c], c0[r] + c1[r]);
      atomicAdd(&sacc[96 + c], c0[r] * c0[r] + c1[r] * c1[r]);
    }
    __syncthreads();
    atomicAdd(&stats_out[t], sacc[t]);
    return;
  }

  // mode 2: bn2 + relu, max over the K=32 neighbors
#pragma unroll
  for (int r = 0; r < 8; r++) {
    int c = m0 + r + 8 * half;
    float v0 = fmaxf(ss2[c] * c0[r] + ss2[96 + c], 0.f);
    float v1 = fmaxf(ss2[c] * c1[r] + ss2[96 + c], 0.f);
    float v = fmaxf(v0, v1);
#pragma unroll
    for (int s = 1; s < 16; s <<= 1) v = fmaxf(v, __shfl_xor(v, s));
    if (l16 == 0) rmax[c] = v;
  }
  __syncthreads();
  if (t < 96) x1out[((size_t)b * 96 + t) * N_ + n] = rmax[t];
}

// =====================================================================
// Offset-attention helpers.  q == k, energy symmetric: e[n,m]=q(:,n).q(:,m)
// =====================================================================
__global__ __launch_bounds__(256) void oa_row_kernel(
    const float* __restrict__ q, float* __restrict__ rowmax,
    float* __restrict__ rowsum) {
  int p = blockIdx.x * 256 + threadIdx.x;
  if (p >= BN_) return;
  int b = p / N_, n = p % N_;
  const float* qb = q + (size_t)b * 24 * N_;
  float qn[24];
#pragma unroll
  for (int c = 0; c < 24; c++) qn[c] = qb[(size_t)c * N_ + n];
  float mx = -3.4e38f, sm = 0.f;
  for (int m = 0; m < N_; m++) {
    float e = 0.f;
#pragma unroll
    for (int c = 0; c < 24; c++) e += qn[c] * qb[(size_t)c * N_ + m];
    if (e > mx) {
      sm = sm * __expf(mx - e) + 1.f;
      mx = e;
    } else {
      sm += __expf(e - mx);
    }
  }
  rowmax[p] = mx;
  rowsum[p] = sm;
}

__global__ __launch_bounds__(256) void oa_col_kernel(
    const float* __restrict__ q, const float* __restrict__ rowmax,
    const float* __restrict__ rowsum, float* __restrict__ cols) {
  int p = blockIdx.x * 256 + threadIdx.x;
  if (p >= BN_) return;
  int b = p / N_, m = p % N_;
  const float* qb = q + (size_t)b * 24 * N_;
  float qm[24];
#pragma unroll
  for (int c = 0; c < 24; c++) qm[c] = qb[(size_t)c * N_ + m];
  float s = 0.f;
  for (int n = 0; n < N_; n++) {
    float e = 0.f;
#pragma unroll
    for (int c = 0; c < 24; c++) e += qm[c] * qb[(size_t)c * N_ + n];
    s += __expf(e - rowmax[b * N_ + n]) / rowsum[b * N_ + n];
  }
  cols[p] = s;
}

// x_r = v @ att2 with att2 tiles regenerated in LDS (no NxN buffer).
// block = 256 (8 waves); waves 0..5 accumulate the 96x16 output tile.
// grid: (NT16, B)
__global__ __launch_bounds__(256) void oa_xr_kernel(
    const float* __restrict__ q, const float* __restrict__ v,
    const float* __restrict__ rowmax, const float* __restrict__ rowsum,
    const float* __restrict__ cols, float* __restrict__ xr) {
  __shared__ float qm[24][16];
  __shared__ float att[64][16];
  __shared__ float invc[16];
  int b = blockIdx.y;
  int m0 = blockIdx.x * 16;
  int t = threadIdx.x;
  const float* qb = q + (size_t)b * 24 * N_;
  const float* vb = v + (size_t)b * 96 * N_;
  if (t < 24 * 16) {
    int c = t / 16, ml = t % 16;
    int m = m0 + ml;
    int mr = (m < N_) ? m : N_ - 1;
    qm[c][ml] = qb[(size_t)c * N_ + mr];
  }
  if (t < 16) {
    int m = m0 + t;
    int mr = (m < N_) ? m : N_ - 1;
    invc[t] = 1.f / (1e-9f + cols[b * N_ + mr]);
  }
  int wave = t >> 5, lane = t & 31, half = lane >> 4, l16 = lane & 15;
  v8f acc = {};
  for (int nb = 0; nb < N_; nb += 64) {
    __syncthreads();
    for (int e = t; e < 64 * 16; e += 256) {
      int nl = e >> 4, ml = e & 15;
      int n = nb + nl;
      float a = 0.f;
      if (n < N_) {
        float ee = 0.f;
#pragma unroll
        for (int c = 0; c < 24; c++) ee += qb[(size_t)c * N_ + n] * qm[c][ml];
        a = __expf(ee - rowmax[b * N_ + n]) / rowsum[b * N_ + n] * invc[ml];
      }
      att[nl][ml] = a;
    }
    __syncthreads();
    if (wave < 6) {
      int c0 = wave * 16;
      for (int kk = 0; kk < 64; kk += 4) {
        int n0v = nb + kk + 2 * half;  // att rows are zero past N_
        int nr0 = min(n0v, N_ - 1), nr1 = min(n0v + 1, N_ - 1);
        v2f a, bb;
        a.x = vb[(size_t)(c0 + l16) * N_ + nr0];
        a.y = vb[(size_t)(c0 + l16) * N_ + nr1];
        bb.x = att[kk + 2 * half][l16];
        bb.y = att[kk + 2 * half + 1][l16];
        acc = wmma_f32(a, bb, acc);
      }
    }
  }
  if (wave < 6) {
    int n = m0 + l16;
    if (n < N_) {
#pragma unroll
      for (int r = 0; r < 8; r++) {
        int c = wave * 16 + r + 8 * half;
        xr[((size_t)b * 96 + c) * N_ + n] = acc[r];
      }
    }
  }
}

__global__ __launch_bounds__(256) void sub_kernel(const float* __restrict__ a,
                                                  const float* __restrict__ bb,
                                                  float* __restrict__ out,
                                                  size_t total) {
  size_t i = (size_t)blockIdx.x * 256 + threadIdx.x;
  if (i < total) out[i] = a[i] - bb[i];
}

// max over N per (b, channel) row; grid = B_*480
__global__ __launch_bounds__(256) void ymax_kernel(const float* __restrict__ y,
                                                   float* __restrict__ ym) {
  int bc = blockIdx.x;
  const float* row = y + (size_t)bc * N_;
  float m = -3.4e38f;
  for (int n = threadIdx.x; n < N_; n += 256) m = fmaxf(m, row[n]);
  __shared__ float red[256];
  red[threadIdx.x] = m;
  __syncthreads();
  for (int o = 128; o > 0; o >>= 1) {
    if (threadIdx.x < o)
      red[threadIdx.x] = fmaxf(red[threadIdx.x], red[threadIdx.x + o]);
    __syncthreads();
  }
  if (threadIdx.x == 0) ym[bc] = red[0];
}

// =====================================================================
// Host orchestration
// =====================================================================
extern "C" void kernel_launch(void* const* d_in, const int* in_sizes, int n_in,
                              void* d_out, int out_size, void* d_ws,
                              size_t ws_size, hipStream_t stream) {
  (void)in_sizes; (void)n_in; (void)out_size; (void)ws_size;
  const float* X = (const float*)d_in[0];
  const float* conv1_w = (const float*)d_in[1];
  const float* bn1_g = (const float*)d_in[2];
  const float* bn1_b = (const float*)d_in[3];
  const float* conv2_w = (const float*)d_in[4];
  const float* bn2_g = (const float*)d_in[5];
  const float* bn2_b = (const float*)d_in[6];
  const float* sg_w1 = (const float*)d_in[7];
  const float* sg_bn1_g = (const float*)d_in[8];
  const float* sg_bn1_b = (const float*)d_in[9];
  const float* sg_w2 = (const float*)d_in[10];
  const float* sg_bn2_g = (const float*)d_in[11];
  const float* sg_bn2_b = (const float*)d_in[12];
  // oa params start at 13, 7 per block
  const float* lin_w = (const float*)d_in[41];
  const float* lin_bn_g = (const float*)d_in[42];
  const float* lin_bn_b = (const float*)d_in[43];

  float* ws = (float*)d_ws;
  float* h48 = ws + W_H48;
  float* x0 = ws + W_X0;
  int* idxb = (int*)(ws + W_IDX);
  float* xbuf[5] = {ws + W_X1, ws + W_X2, ws + W_X3, ws + W_X4, ws + W_X5};
  float* qbuf = ws + W_Q;
  float* vbuf = ws + W_V;
  float* xrbuf = ws + W_XR;
  float* tbuf = ws + W_T;
  float* rmax = ws + W_RMAX;
  float* rsum = ws + W_RSUM;
  float* colsum = ws + W_COLS;
  float* stats = ws + W_STATS;
  float* yout = (float*)d_out;  // (B,480,N) then (B,480) tail

  const float invBN = 1.f / (float)BN_;
  const float invBNK = 1.f / (float)(BN_ * KNN);

  hipMemsetAsync(stats, 0, STATS_FLOATS * sizeof(float), stream);

  // ---- stage 1: conv1 -> bn1 -> relu -> conv2 -> bn2 -> relu = x0
  conv1_kernel<<<dim3((BN_ + 255) / 256, 48), 256, 0, stream>>>(X, conv1_w,
                                                                h48);
  stats_kernel<<<48, 256, 0, stream>>>(h48, stats + S_BN1, 48);
  bn_finalize_kernel<<<1, 64, 0, stream>>>(stats + S_BN1, bn1_g, bn1_b, 48,
                                           invBN);
  bn_apply_kernel<<<(B_ * 48 * N_ + 255) / 256, 256, 0, stream>>>(
      h48, nullptr, h48, stats + S_BN1, 48, 0);
  gemm_wmma_kernel<<<dim3(NT64, 3, B_), 32, 0, stream>>>(conv2_w, nullptr, h48,
                                                         x0, 48, 48);
  stats_kernel<<<48, 256, 0, stream>>>(x0, stats + S_BN2, 48);
  bn_finalize_kernel<<<1, 64, 0, stream>>>(stats + S_BN2, bn2_g, bn2_b, 48,
                                           invBN);
  bn_apply_kernel<<<(B_ * 48 * N_ + 255) / 256, 256, 0, stream>>>(
      x0, nullptr, x0, stats + S_BN2, 48, 0);

  // ---- kNN
  knn_kernel<<<(BN_ + 63) / 64, 64, 0, stream>>>(X, idxb);

  // ---- sample-and-group MLP (3 recompute passes, BN in training mode)
  sg_kernel<<<BN_, 192, 0, stream>>>(x0, idxb, sg_w1, sg_w2, nullptr, nullptr,
                                     stats + S_SG1, nullptr, 0);
  bn_finalize_kernel<<<2, 64, 0, stream>>>(stats + S_SG1, sg_bn1_g, sg_bn1_b,
                                           96, invBNK);
  sg_kernel<<<BN_, 192, 0, stream>>>(x0, idxb, sg_w1, sg_w2, stats + S_SG1,
                                     nullptr, stats + S_SG2, nullptr, 1);
  bn_finalize_kernel<<<2, 64, 0, stream>>>(stats + S_SG2, sg_bn2_g, sg_bn2_b,
                                           96, invBNK);
  sg_kernel<<<BN_, 192, 0, stream>>>(x0, idxb, sg_w1, sg_w2, stats + S_SG1,
                                     stats + S_SG2, nullptr, xbuf[0], 2);

  // ---- 4 offset-attention blocks
  const int sslot[4] = {S_T0, S_T1, S_T2, S_T3};
  for (int i = 0; i < 4; i++) {
    const float* qk_w = (const float*)d_in[13 + 7 * i + 0];
    const float* v_w = (const float*)d_in[13 + 7 * i + 1];
    const float* v_b = (const float*)d_in[13 + 7 * i + 2];
    const float* t_w = (const float*)d_in[13 + 7 * i + 3];
    const float* t_b = (const float*)d_in[13 + 7 * i + 4];
    const float* obn_g = (const float*)d_in[13 + 7 * i + 5];
    const float* obn_b = (const float*)d_in[13 + 7 * i + 6];
    const float* xin = xbuf[i];
    float* xout = xbuf[i + 1];
    float* st = stats + sslot[i];

    gemm_wmma_kernel<<<dim3(NT64, 2, B_), 32, 0, stream>>>(qk_w, nullptr, xin,
                                                           qbuf, 24, 96);
    gemm_wmma_kernel<<<dim3(NT64, 6, B_), 32, 0, stream>>>(v_w, v_b, xin, vbuf,
                                                           96, 96);
    oa_row_kernel<<<(BN_ + 255) / 256, 256, 0, stream>>>(qbuf, rmax, rsum);
    oa_col_kernel<<<(BN_ + 255) / 256, 256, 0, stream>>>(qbuf, rmax, rsum,
                                                         colsum);
    oa_xr_kernel<<<dim3(NT16, B_), 256, 0, stream>>>(qbuf, vbuf, rmax, rsum,
                                                     colsum, xrbuf);
    sub_kernel<<<(B_ * 96 * N_ + 255) / 256, 256, 0, stream>>>(
        xin, xrbuf, vbuf, (size_t)B_ * 96 * N_);  // vbuf := x - x_r
    gemm_wmma_kernel<<<dim3(NT64, 6, B_), 32, 0, stream>>>(t_w, t_b, vbuf, tbuf,
                                                           96, 96);
    stats_kernel<<<96, 256, 0, stream>>>(tbuf, st, 96);
    bn_finalize_kernel<<<2, 64, 0, stream>>>(st, obn_g, obn_b, 96, invBN);
    bn_apply_kernel<<<(B_ * 96 * N_ + 255) / 256, 256, 0, stream>>>(
        tbuf, xin, xout, st, 96, 1);  // xout = xin + relu(bn(t))
  }

  // ---- lin: concat(x1..x5) -> 480x480 GEMM -> bn -> leaky relu; + row max
  gemm_cat_kernel<<<dim3(NT64, 30, B_), 32, 0, stream>>>(
      lin_w, xbuf[0], xbuf[1], xbuf[2], xbuf[3], xbuf[4], yout);
  stats_kernel<<<480, 256, 0, stream>>>(yout, stats + S_LIN, 480);
  bn_finalize_kernel<<<8, 64, 0, stream>>>(stats + S_LIN, lin_bn_g, lin_bn_b,
                                           480, invBN);
  bn_apply_kernel<<<(B_ * 480 * N_ + 255) / 256, 256, 0, stream>>>(
      yout, nullptr, yout, stats + S_LIN, 480, 2);
  ymax_kernel<<<B_ * 480, 256, 0, stream>>>(yout,
                                            yout + (size_t)B_ * 480 * N_);
}